// ModifiedSinkhorn_86844238725720
// MI455X (gfx1250) — compile-verified
//
#include <hip/hip_runtime.h>

// ---------------------------------------------------------------------------
// Sinkhorn divergence (3x) for MI455X / gfx1250.
//
// Base-2 log domain:  w_ij = t2[j] + K2*dot(x_i,y_j)  (row const pulled out)
//   t2[j] = log2(w_j) + K2*(pot_j - h_j),  h = 0.5*||x||^2, K2 = log2(e)/eps
//   softmin_i = h_i - eps*ln2*(m_i + log2(s_i)),  (m,s) = max / exp2-sum
//
// w tiles come from v_wmma_f32_16x16x4_f32 with the affine trick:
//   A (16x4) = (K2*x0, K2*x1, 1, 0) rows, B (4x16) = (y0, y1, t2_j, 0), C = 0.
// Column data is packed as float2 in LDS: pack[j]={y0,y1}, pack[256+j]={t2,0},
// so the B fragment is ONE unconditional ds_load_b64 per tile (no exec masks).
// ---------------------------------------------------------------------------

typedef float v2f __attribute__((ext_vector_type(2)));
typedef float v8f __attribute__((ext_vector_type(8)));

#define SK_N     256
#define SK_ITERS 20
#define SK_K2    577.0780163555852f      // log2(e)/eps, eps = 0.05^2
#define SK_C2    0.0017328679513998632f  // eps*ln2
#define SK_NEG   -1.0e30f

static __device__ __forceinline__ float fast_exp2(float x) {
#if __has_builtin(__builtin_amdgcn_exp2f)
  return __builtin_amdgcn_exp2f(x);      // v_exp_f32 (base-2)
#else
  return exp2f(x);
#endif
}
static __device__ __forceinline__ float fast_log2(float x) {
#if __has_builtin(__builtin_amdgcn_logf)
  return __builtin_amdgcn_logf(x);       // v_log_f32 (base-2)
#else
  return log2f(x);
#endif
}

static __device__ __forceinline__ float xmax16(float v) {
  v = fmaxf(v, __shfl_xor(v, 1, 32));
  v = fmaxf(v, __shfl_xor(v, 2, 32));
  v = fmaxf(v, __shfl_xor(v, 4, 32));
  v = fmaxf(v, __shfl_xor(v, 8, 32));
  return v;
}
static __device__ __forceinline__ float xsum16(float v) {
  v += __shfl_xor(v, 1, 32);
  v += __shfl_xor(v, 2, 32);
  v += __shfl_xor(v, 4, 32);
  v += __shfl_xor(v, 8, 32);
  return v;
}

// One softmin over the implicit 256x256 cost matrix.
// rpack: row points {x0,x1} in [0..255]; rh: 0.5*||x||^2 per row.
// cpack: col points in [0..255], {t2,0} in [256..511] (prebuilt this pass).
// mode 0: out[row] = r ; mode 1: out[row] = 0.5*(out[row] + r)
static __device__ __forceinline__ void softmin_tiles(
    const v2f* rpack, const float* rh, const v2f* cpack,
    float* outp, int mode)
{
  const int tid  = threadIdx.x;
  const int lane = tid & 31;
  const int wave = tid >> 5;
  const int l    = lane & 15;
  const bool lo  = lane < 16;
  const int cofs = l + (lo ? 0 : 256);   // lower half: points, upper: t2 slot

  const v8f Cz = {};                     // loop-invariant zero accumulator

  for (int rt = wave * 2; rt < wave * 2 + 2; ++rt) {
    const int rbase = rt * 16;

    // A fragment: M across lanes, K across {VGPR, half}; K=2 is the affine 1.
    const v2f rv = rpack[rbase + l];     // unconditional broadcast load
    v2f A;
    A.x = lo ? SK_K2 * rv.x : 1.0f;      // K=0 | K=2
    A.y = lo ? SK_K2 * rv.y : 0.0f;      // K=1 | K=3

    float m[8], s[8];
#pragma unroll
    for (int v = 0; v < 8; ++v) m[v] = -3.0e38f;

    // ---- sweep 1: per-lane column max over 16 column tiles ----
    for (int ct = 0; ct < 16; ++ct) {
      const v2f bv = cpack[ct * 16 + cofs];          // one ds_load_b64
      v8f D = __builtin_amdgcn_wmma_f32_16x16x4_f32(
          false, A, false, bv, (short)0, Cz, false, false);
#pragma unroll
      for (int v = 0; v < 8; ++v) m[v] = fmaxf(m[v], D[v]);
    }
#pragma unroll
    for (int v = 0; v < 8; ++v) { m[v] = xmax16(m[v]); s[v] = 0.0f; }

    // ---- sweep 2: sum of exp2(w - m) ----
    for (int ct = 0; ct < 16; ++ct) {
      const v2f bv = cpack[ct * 16 + cofs];
      v8f D = __builtin_amdgcn_wmma_f32_16x16x4_f32(
          false, A, false, bv, (short)0, Cz, false, false);
#pragma unroll
      for (int v = 0; v < 8; ++v) s[v] += fast_exp2(D[v] - m[v]);
    }
#pragma unroll
    for (int v = 0; v < 8; ++v) s[v] = xsum16(s[v]);

    // ---- writers: lane v (half0) -> row rbase+v ; lane 16+v -> rbase+8+v ----
#pragma unroll
    for (int v = 0; v < 8; ++v) {
      if (l == v) {
        const int row = rbase + v + (lo ? 0 : 8);
        const float r = rh[row] - SK_C2 * (m[v] + fast_log2(s[v]));
        outp[row] = (mode == 0) ? r : 0.5f * (outp[row] + r);
      }
    }
  }
}

static __device__ __forceinline__ void softmin_pass(
    const v2f* rpack, const float* rh,
    v2f* cpack, const float* ctb, const float* cpot,
    float* outp, int mode)
{
  const int tid = threadIdx.x;
  v2f t;
  t.x = ctb[tid] + SK_K2 * cpot[tid];
  t.y = 0.0f;
  cpack[SK_N + tid] = t;                 // t2 slot (K=2 of B), K=3 stays 0
  __syncthreads();
  softmin_tiles(rpack, rh, cpack, outp, mode);
  __syncthreads();
}

// grid = (B, 9): blockIdx.y = div*3 + part; part 0 = f/g chain, 1 = p, 2 = q
__global__ __launch_bounds__(256, 4) void sinkhorn_part_kernel(
    const float* __restrict__ a_mask, const float* __restrict__ pc_a,
    const float* __restrict__ b_mask, const float* __restrict__ pc_b,
    float* __restrict__ ws)
{
  __shared__ v2f   apack[2 * SK_N], bpack[2 * SK_N];
  __shared__ float ah[SK_N], atb[SK_N], bh[SK_N], btb[SK_N];
  __shared__ float potA[SK_N], potB[SK_N];

  const int b    = blockIdx.x;
  const int div  = blockIdx.y / 3;
  const int part = blockIdx.y % 3;
  const int tid  = threadIdx.x;

  if (part != 2) {  // need A side
    const float* pa = pc_a + ((size_t)b * SK_N + tid) * 3;
    const float msk = a_mask[(size_t)b * SK_N + tid];
    float x0, x1, w;
    if (div == 2) { x0 = msk * pa[2]; x1 = 0.0f; w = msk; }
    else          { x0 = pa[0]; x1 = pa[1]; w = (div == 0) ? msk * pa[2] : msk; }
    const float h = 0.5f * (x0 * x0 + x1 * x1);
    const float lw2 = (w > 0.0f) ? fast_log2(fmaxf(w, 1e-30f)) : SK_NEG;
    v2f xv; xv.x = x0; xv.y = x1;
    apack[tid] = xv; ah[tid] = h;
    atb[tid] = lw2 - SK_K2 * h; potA[tid] = 0.0f;
  }
  if (part != 1) {  // need B side
    const float* pb = pc_b + ((size_t)b * SK_N + tid) * 3;
    const float msk = b_mask[(size_t)b * SK_N + tid];
    float y0, y1, w;
    if (div == 2) { y0 = msk * pb[2]; y1 = 0.0f; w = msk; }
    else          { y0 = pb[0]; y1 = pb[1]; w = (div == 0) ? msk * pb[2] : msk; }
    const float h = 0.5f * (y0 * y0 + y1 * y1);
    const float lw2 = (w > 0.0f) ? fast_log2(fmaxf(w, 1e-30f)) : SK_NEG;
    v2f yv; yv.x = y0; yv.y = y1;
    bpack[tid] = yv; bh[tid] = h;
    btb[tid] = lw2 - SK_K2 * h; potB[tid] = 0.0f;
  }
  __syncthreads();

  const size_t base = (size_t)(b * 3 + div) * 4 * SK_N;

  if (part == 0) {           // f/g chain
    for (int it = 0; it < SK_ITERS; ++it) {
      softmin_pass(apack, ah, bpack, btb, potB, potA, 0); // f <- lse over g
      softmin_pass(bpack, bh, apack, atb, potA, potB, 0); // g <- lse over f
    }
    // final f uses g_20; final g == g_20 (exact recompute in reference)
    softmin_pass(apack, ah, bpack, btb, potB, potA, 0);
    ws[base + tid]        = potA[tid];   // f
    ws[base + SK_N + tid] = potB[tid];   // g
  } else if (part == 1) {    // p chain (self, A side)
    for (int it = 0; it < SK_ITERS; ++it)
      softmin_pass(apack, ah, apack, atb, potA, potA, 1);
    softmin_pass(apack, ah, apack, atb, potA, potA, 0);  // final: no averaging
    ws[base + 2 * SK_N + tid] = potA[tid];  // p
  } else {                   // q chain (self, B side)
    for (int it = 0; it < SK_ITERS; ++it)
      softmin_pass(bpack, bh, bpack, btb, potB, potB, 1);
    softmin_pass(bpack, bh, bpack, btb, potB, potB, 0);
    ws[base + 3 * SK_N + tid] = potB[tid];  // q
  }
}

// Deterministic combine: loss[b] = sum_div [ sum_i aw*(f-p) + sum_j bw*(g-q) ]
__global__ __launch_bounds__(256) void sinkhorn_combine_kernel(
    const float* __restrict__ a_mask, const float* __restrict__ pc_a,
    const float* __restrict__ b_mask, const float* __restrict__ pc_b,
    const float* __restrict__ ws, float* __restrict__ out)
{
  __shared__ float red[SK_N];
  const int b = blockIdx.x;
  const int tid = threadIdx.x;

  const float am  = a_mask[(size_t)b * SK_N + tid];
  const float bm  = b_mask[(size_t)b * SK_N + tid];
  const float apt = am * pc_a[((size_t)b * SK_N + tid) * 3 + 2];
  const float bpt = bm * pc_b[((size_t)b * SK_N + tid) * 3 + 2];

  float acc = 0.0f;
  for (int div = 0; div < 3; ++div) {
    const size_t base = (size_t)(b * 3 + div) * 4 * SK_N;
    const float f = ws[base + tid];
    const float g = ws[base + SK_N + tid];
    const float p = ws[base + 2 * SK_N + tid];
    const float q = ws[base + 3 * SK_N + tid];
    const float aw = (div == 0) ? apt : am;
    const float bw = (div == 0) ? bpt : bm;
    acc += aw * (f - p) + bw * (g - q);
  }
  red[tid] = acc;
  __syncthreads();
  for (int s = 128; s > 0; s >>= 1) {
    if (tid < s) red[tid] += red[tid + s];
    __syncthreads();
  }
  if (tid == 0) out[b] = red[0];
}

extern "C" void kernel_launch(void* const* d_in, const int* in_sizes, int n_in,
                              void* d_out, int out_size, void* d_ws, size_t ws_size,
                              hipStream_t stream) {
  const float* a_mask = (const float*)d_in[0];
  const float* pc_a   = (const float*)d_in[1];
  const float* b_mask = (const float*)d_in[2];
  const float* pc_b   = (const float*)d_in[3];
  float* ws  = (float*)d_ws;
  float* out = (float*)d_out;

  const int B = in_sizes[0] / SK_N;  // 128

  dim3 grid1(B, 9);
  sinkhorn_part_kernel<<<grid1, 256, 0, stream>>>(a_mask, pc_a, b_mask, pc_b, ws);
  sinkhorn_combine_kernel<<<B, 256, 0, stream>>>(a_mask, pc_a, b_mask, pc_b, ws, out);
}